// AGNNConv_5866925326657
// MI455X (gfx1250) — compile-verified
//
#include <hip/hip_runtime.h>
#include <math.h>

typedef float v2f __attribute__((ext_vector_type(2)));
typedef float v8f __attribute__((ext_vector_type(8)));

// ---- monotonic float<->uint key (total order under unsigned compare) ----
__device__ __forceinline__ unsigned fkey(float f) {
    unsigned b = __float_as_uint(f);
    return (b & 0x80000000u) ? ~b : (b | 0x80000000u);
}
__device__ __forceinline__ float funkey(unsigned k) {
    unsigned b = (k & 0x80000000u) ? (k ^ 0x80000000u) : ~k;
    return __uint_as_float(b);
}
#define FKEY_NEG_INF 0x007FFFFFu  // fkey(-inf)

__device__ __forceinline__ void atom_add_f32(float* p, float v) {
    __hip_atomic_fetch_add(p, v, __ATOMIC_RELAXED, __HIP_MEMORY_SCOPE_AGENT);
}
__device__ __forceinline__ void atom_max_u32(unsigned* p, unsigned v) {
    __hip_atomic_fetch_max(p, v, __ATOMIC_RELAXED, __HIP_MEMORY_SCOPE_AGENT);
}

// ---------------------------------------------------------------------------
// Kernel 1: inverse L2 norm per node via WMMA Gram diagonal.
// One wave handles 16 rows. A = 16x4 f32 chunk of feat; the f32 16x16x4 A and
// B VGPR layouts mirror each other, so wmma(a, a, acc) accumulates F_blk*F_blkT.
// Diagonal of the 16x16 result = squared row norms.
// ---------------------------------------------------------------------------
__global__ void agnn_norm_wmma(const float* __restrict__ feat,
                               float* __restrict__ inv_norm, int n) {
    const int tid  = blockIdx.x * blockDim.x + threadIdx.x;
    const int wid  = tid >> 5;
    const int lane = threadIdx.x & 31;
    const int base = wid * 16;
    if (base >= n) return;                 // wave-uniform exit: EXEC stays all-ones

    int row = base + (lane & 15);
    if (row >= n) row = n - 1;             // clamp (keeps EXEC full for WMMA)

#if __has_builtin(__builtin_amdgcn_wmma_f32_16x16x4_f32)
    // lane<16 holds K = {0,1} of each 4-chunk, lane>=16 holds K = {2,3}
    const float* rp = feat + (size_t)row * 64 + ((lane >> 4) << 1);
    v8f acc = {};
#pragma unroll
    for (int c = 0; c < 16; ++c) {
        v2f a = *(const v2f*)(rp + 4 * c);
        acc = __builtin_amdgcn_wmma_f32_16x16x4_f32(
            false, a, false, a, (short)0, acc, false, false);
    }
    // Diagonal extraction: D vgpr r holds M=r (lanes 0-15, N=lane) and
    // M=8+r (lanes 16-31, N=lane-16). diag(m): m<8 -> acc[m]@lane m;
    // m>=8 -> acc[m-8]@lane m+16.
    float dsq = 0.0f;
#pragma unroll
    for (int r = 0; r < 8; ++r) {
        if (lane == r)      dsq = acc[r];
        if (lane == 24 + r) dsq = acc[r];
    }
    const int m = (lane < 8) ? lane : ((lane >= 24) ? (lane - 16) : -1);
    if (m >= 0 && base + m < n)
        inv_norm[base + m] = 1.0f / fmaxf(sqrtf(dsq), 1e-12f);
#else
    // VALU fallback: lane pair (l, l+16) splits a row's 64 elements.
    const float* rp = feat + (size_t)row * 64 + ((lane >> 4) << 5);
    float s = 0.0f;
#pragma unroll
    for (int i = 0; i < 32; ++i) { float v = rp[i]; s += v * v; }
    s += __shfl_xor(s, 16, 32);
    if (lane < 16 && base + lane < n)
        inv_norm[base + lane] = 1.0f / fmaxf(sqrtf(s), 1e-12f);
#endif
}

// ---------------------------------------------------------------------------
// Kernel 2: init segment buffers (emax keys to key(-inf), denom to 0)
// ---------------------------------------------------------------------------
__global__ void agnn_seg_init(unsigned* __restrict__ emax_key,
                              float* __restrict__ denom, int n) {
    int i = blockIdx.x * blockDim.x + threadIdx.x;
    if (i < n) { emax_key[i] = FKEY_NEG_INF; denom[i] = 0.0f; }
}

// ---------------------------------------------------------------------------
// Kernel 3: out = (1+eps)*feat   (float4 vectorized; fully rewrites d_out)
// ---------------------------------------------------------------------------
__global__ void agnn_scale_out(const float* __restrict__ feat,
                               const float* __restrict__ eps,
                               float* __restrict__ out, int total4) {
    int i = blockIdx.x * blockDim.x + threadIdx.x;
    if (i >= total4) return;
    const float s = 1.0f + eps[0];
    float4 f = ((const float4*)feat)[i];
    float4 o; o.x = s * f.x; o.y = s * f.y; o.z = s * f.z; o.w = s * f.w;
    ((float4*)out)[i] = o;
}

// ---------------------------------------------------------------------------
// Kernel 4: segment max of e = beta*w over src
// ---------------------------------------------------------------------------
__global__ void agnn_edge_max(const float* __restrict__ ew,
                              const float* __restrict__ beta,
                              const int* __restrict__ src,
                              unsigned* __restrict__ emax_key, int E) {
    int e = blockIdx.x * blockDim.x + threadIdx.x;
    if (e >= E) return;
    __builtin_prefetch(src + e + 8192, 0, 1);   // global_prefetch_b8
    __builtin_prefetch(ew  + e + 8192, 0, 1);
    const float w = beta[0] * ew[e];
    atom_max_u32(&emax_key[src[e]], fkey(w));
}

// ---------------------------------------------------------------------------
// Kernel 5: e_exp = exp(e - emax[src]); denom[src] += e_exp
// ---------------------------------------------------------------------------
__global__ void agnn_edge_sum(const float* __restrict__ ew,
                              const float* __restrict__ beta,
                              const int* __restrict__ src,
                              const unsigned* __restrict__ emax_key,
                              float* __restrict__ denom,
                              float* __restrict__ eexp, int E) {
    int e = blockIdx.x * blockDim.x + threadIdx.x;
    if (e >= E) return;
    __builtin_prefetch(src + e + 8192, 0, 1);
    __builtin_prefetch(ew  + e + 8192, 0, 1);
    const int   s  = src[e];
    const float w  = beta[0] * ew[e];
    const float ex = expf(w - funkey(emax_key[s]));
    eexp[e] = ex;
    atom_add_f32(&denom[s], ex);
}

// ---------------------------------------------------------------------------
// Kernel 6: wave-per-edge scatter. 32 lanes x float2 = 64-wide row.
// out[dst] += feat[src] * inv_norm[src] * (eexp/denom[src])
// ---------------------------------------------------------------------------
__global__ void agnn_scatter(const float* __restrict__ feat,
                             const int* __restrict__ src,
                             const int* __restrict__ dst,
                             const float* __restrict__ eexp,
                             const float* __restrict__ denom,
                             const float* __restrict__ inv_norm,
                             float* __restrict__ out, int E) {
    const int tid  = blockIdx.x * blockDim.x + threadIdx.x;
    const int e    = tid >> 5;
    const int lane = tid & 31;
    if (e >= E) return;
    const int s = src[e];
    const int d = dst[e];
    const float scale = (eexp[e] / denom[s]) * inv_norm[s];
    const v2f v = ((const v2f*)(feat + (size_t)s * 64))[lane];
    float* op = out + (size_t)d * 64 + 2 * lane;
    atom_add_f32(op,     v.x * scale);
    atom_add_f32(op + 1, v.y * scale);
}

// ---------------------------------------------------------------------------
extern "C" void kernel_launch(void* const* d_in, const int* in_sizes, int n_in,
                              void* d_out, int out_size, void* d_ws, size_t ws_size,
                              hipStream_t stream) {
    const float* feat = (const float*)d_in[0];
    const float* ew   = (const float*)d_in[1];
    const float* beta = (const float*)d_in[2];
    const float* eps  = (const float*)d_in[3];
    const int*   src  = (const int*)d_in[4];
    const int*   dst  = (const int*)d_in[5];
    float*       out  = (float*)d_out;

    const int N = in_sizes[0] / 64;
    const int E = in_sizes[1];

    // workspace layout: inv_norm[N] | emax_key[N] | denom[N] | eexp[E]
    float*    inv_norm = (float*)d_ws;
    unsigned* emax_key = (unsigned*)(inv_norm + N);
    float*    denom    = (float*)(emax_key + N);
    float*    eexp     = denom + N;

    const int BLK = 256;
    const int waves       = (N + 15) / 16;
    const int norm_blocks = (waves * 32 + BLK - 1) / BLK;
    const int n_blocks    = (N + BLK - 1) / BLK;
    const int total4      = N * 16;                       // N*64/4
    const int sc_blocks   = (total4 + BLK - 1) / BLK;
    const int e_blocks    = (E + BLK - 1) / BLK;
    const long long sct   = (long long)E * 32;
    const int scat_blocks = (int)((sct + BLK - 1) / BLK);

    agnn_norm_wmma<<<norm_blocks, BLK, 0, stream>>>(feat, inv_norm, N);
    agnn_seg_init <<<n_blocks,    BLK, 0, stream>>>(emax_key, denom, N);
    agnn_scale_out<<<sc_blocks,   BLK, 0, stream>>>(feat, eps, out, total4);
    agnn_edge_max <<<e_blocks,    BLK, 0, stream>>>(ew, beta, src, emax_key, E);
    agnn_edge_sum <<<e_blocks,    BLK, 0, stream>>>(ew, beta, src, emax_key, denom, eexp, E);
    agnn_scatter  <<<scat_blocks, BLK, 0, stream>>>(feat, src, dst, eexp, denom, inv_norm, out, E);
}